// ProposalLayer_46832323396033
// MI455X (gfx1250) — compile-verified
//
#include <hip/hip_runtime.h>
#include <math.h>

// ---------------- problem constants (match reference) ----------------
#define AA 9
constexpr int Hh   = 50;
constexpr int Ww   = 80;
constexpr int N    = Hh * Ww * AA;   // 36000 anchors
constexpr int CHUNK  = 1024;         // floats per async-staged score chunk (4 KB)
constexpr int NP   = 36 * CHUNK;     // 36864: N padded to chunk multiple
constexpr int K    = 6000;           // PRE_NMS
constexpr int POST = 300;            // POST_NMS
constexpr float NMS_THRESH = 0.7f;
constexpr float MIN_SIZE   = 16.0f;

// Base anchors from generate_anchors(16) computed exactly (numpy banker's
// rounding: ws=[23,16,11], hs=[12,16,22]), order: ratio-major, scale-minor.
__constant__ float c_base_anchors[AA * 4] = {
    -84.f,  -40.f,   99.f,   55.f,
   -176.f,  -88.f,  191.f,  103.f,
   -360.f, -184.f,  375.f,  199.f,
    -56.f,  -56.f,   71.f,   71.f,
   -120.f, -120.f,  135.f,  135.f,
   -248.f, -248.f,  263.f,  263.f,
    -36.f,  -80.f,   51.f,   95.f,
    -80.f, -168.f,   95.f,  183.f,
   -168.f, -344.f,  183.f,  359.f,
};

// ---- LDS byte-offset of a shared-memory pointer (for async/TDM LDS dests) ----
__device__ __forceinline__ unsigned ldsOffset(const void* p) {
  // addrspacecast generic->LDS, then ptrtoint gives the 32-bit LDS offset.
  return (unsigned)(unsigned long long)(const __attribute__((address_space(3))) char*)p;
}

// Issue one per-lane 16B async copy global->LDS (GVS mode: saddr + 32b voffset).
__device__ __forceinline__ void asyncLoadB128(unsigned ldsOff, unsigned gByteOff,
                                              const void* base) {
  asm volatile("global_load_async_to_lds_b128 %0, %1, %2"
               :: "v"(ldsOff), "v"(gByteOff), "s"(base) : "memory");
}
__device__ __forceinline__ void waitAsync0() {
  asm volatile("s_wait_asynccnt 0" ::: "memory");
}
__device__ __forceinline__ void waitAsync1() {
  asm volatile("s_wait_asynccnt 1" ::: "memory");
}

// ---- Tensor Data Mover: 1-D bulk DMA global -> LDS (TENSORcnt tracked) ----
typedef unsigned int u32x4 __attribute__((ext_vector_type(4)));
typedef int          i32x4 __attribute__((ext_vector_type(4)));
typedef int          i32x8 __attribute__((ext_vector_type(8)));

// DMA `nElem` dwords from `gsrc` into LDS at byte offset `ldsAddr`.
// D# per CDNA5 ISA ch.8: group0 = {count=1, lds_addr, global_addr, type=2},
// group1 = {mask=0, data_size=4B, tensor_dim0=nElem, tensor_dim1=1,
//           tile_dim0=nElem, tile_dim1=1, dim0_stride=nElem}, groups 2/3 = 0.
__device__ __forceinline__ void tdmLoad1D(unsigned ldsAddr, const void* gsrc,
                                          unsigned nElem) {
  const unsigned long long ga = (unsigned long long)gsrc;
  u32x4 g0;
  g0.x = 1u;                                       // count=1, user descriptor
  g0.y = ldsAddr;                                  // lds_addr [63:32]
  g0.z = (unsigned)ga;                             // global_addr [95:64]
  g0.w = (unsigned)(ga >> 32) | (2u << 30);        // global_addr hi | type=2
  i32x8 g1;
  g1[0] = (int)(2u << 16);                         // data_size=2 (4 bytes)
  g1[1] = (int)((nElem & 0xFFFFu) << 16);          // tensor_dim0[15:0] @ [63:48]
  g1[2] = (int)((nElem >> 16) | (1u << 16));       // dim0 hi | tensor_dim1=1
  g1[3] = (int)((nElem & 0xFFFFu) << 16);          // tile_dim0 @ [127:112]
  g1[4] = 1;                                       // tile_dim1=1, tile_dim2=0
  g1[5] = (int)nElem;                              // tensor_dim0_stride lo
  g1[6] = 0;                                       // stride hi, dim1_stride lo
  g1[7] = 0;
  const i32x4 z4 = {0, 0, 0, 0};
#if __clang_major__ >= 23
  const i32x8 z8 = {0, 0, 0, 0, 0, 0, 0, 0};
  __builtin_amdgcn_tensor_load_to_lds(g0, g1, z4, z4, z8, 0);
#else
  __builtin_amdgcn_tensor_load_to_lds(g0, g1, z4, z4, 0);
#endif
}

// ---------------------------------------------------------------------------
// Kernel 1: anchor decode + clip + min-size mask. Writes boxes[N*4] and
// scores[NP] (pad filled with -inf so the rank kernel's padded tail is inert).
// ---------------------------------------------------------------------------
__global__ __launch_bounds__(256)
void decode_kernel(const float* __restrict__ scores,
                   const float* __restrict__ deltas,
                   const float* __restrict__ im_info,
                   float* __restrict__ boxesAll,
                   float* __restrict__ scoresAll) {
  int n = blockIdx.x * 256 + threadIdx.x;
  if (n >= NP) return;
  if (n >= N) { scoresAll[n] = -INFINITY; return; }

  const int a  = n % AA;
  const int hw = n / AA;          // h*W + w
  const int w  = hw % Ww;
  const int h  = hw / Ww;
  const float sx = (float)w * 16.0f;
  const float sy = (float)h * 16.0f;

  const float x1 = c_base_anchors[a * 4 + 0] + sx;
  const float y1 = c_base_anchors[a * 4 + 1] + sy;
  const float x2 = c_base_anchors[a * 4 + 2] + sx;
  const float y2 = c_base_anchors[a * 4 + 3] + sy;

  // (h,w,a)-ordered gathers from NCHW tensors.
  const int HW = Hh * Ww;
  const float sc = scores[(AA + a) * HW + hw];
  const float dx = deltas[(4 * a + 0) * HW + hw];
  const float dy = deltas[(4 * a + 1) * HW + hw];
  float dw = deltas[(4 * a + 2) * HW + hw];
  float dh = deltas[(4 * a + 3) * HW + hw];
  dw = fminf(fmaxf(dw, -10.0f), 10.0f);
  dh = fminf(fmaxf(dh, -10.0f), 10.0f);

  const float aw  = x2 - x1 + 1.0f;
  const float ah  = y2 - y1 + 1.0f;
  const float acx = x1 + 0.5f * aw;
  const float acy = y1 + 0.5f * ah;

  const float pcx = dx * aw + acx;
  const float pcy = dy * ah + acy;
  const float pw  = expf(dw) * aw;
  const float ph  = expf(dh) * ah;

  const float im_h = im_info[0], im_w = im_info[1], im_s = im_info[2];
  float bx1 = fminf(fmaxf(pcx - 0.5f * pw, 0.0f), im_w - 1.0f);
  float by1 = fminf(fmaxf(pcy - 0.5f * ph, 0.0f), im_h - 1.0f);
  float bx2 = fminf(fmaxf(pcx + 0.5f * pw, 0.0f), im_w - 1.0f);
  float by2 = fminf(fmaxf(pcy + 0.5f * ph, 0.0f), im_h - 1.0f);

  boxesAll[n * 4 + 0] = bx1;
  boxesAll[n * 4 + 1] = by1;
  boxesAll[n * 4 + 2] = bx2;
  boxesAll[n * 4 + 3] = by2;

  const float min_sz = MIN_SIZE * im_s;
  const bool valid = ((bx2 - bx1 + 1.0f) >= min_sz) && ((by2 - by1 + 1.0f) >= min_sz);
  scoresAll[n] = valid ? sc : -INFINITY;
}

// ---------------------------------------------------------------------------
// Kernel 2: exact stable top-K via rank counting.
// rank_i = #{j : s_j > s_i} + #{j : s_j == s_i && j < i}  (== lax.top_k order).
// Each block streams the whole padded score vector through LDS using
// double-buffered async global->LDS B128 copies (ASYNCcnt).
// ---------------------------------------------------------------------------
__global__ __launch_bounds__(256)
void rank_kernel(const float* __restrict__ scoresAll,
                 const float* __restrict__ boxesAll,
                 float* __restrict__ topScores,
                 float* __restrict__ topBoxes) {
  __shared__ float sh[2][CHUNK];

  const int tid = threadIdx.x;
  const int i   = blockIdx.x * 256 + tid;
  const bool act = (i < N);
  const float si = act ? scoresAll[i] : 0.0f;
  int rank = 0;

  const unsigned lds0  = ldsOffset(&sh[0][tid * 4]);
  const unsigned lds1  = ldsOffset(&sh[1][tid * 4]);
  const unsigned gbyte = (unsigned)(tid * 16);   // 16 B per lane, 256 lanes = 4 KB

  constexpr int NCHUNK = NP / CHUNK;             // 36
  asyncLoadB128(lds0, gbyte, scoresAll);         // prefetch chunk 0
  for (int c = 0; c < NCHUNK; ++c) {
    if (c + 1 < NCHUNK) {                        // prefetch next chunk, keep 1 in flight
      asyncLoadB128(((c + 1) & 1) ? lds1 : lds0,
                    gbyte + (unsigned)((c + 1) * CHUNK * 4), scoresAll);
      waitAsync1();                              // chunk c complete (done in order)
    } else {
      waitAsync0();
    }
    __syncthreads();
    if (act) {
      const float4* sh4 = (const float4*)sh[c & 1];
      const int base = c * CHUNK;
#pragma unroll 4
      for (int j4 = 0; j4 < CHUNK / 4; ++j4) {
        const float4 v = sh4[j4];                // LDS b128 broadcast read
        const int jj = base + j4 * 4;
        rank += (v.x > si) || (v.x == si && (jj + 0) < i);
        rank += (v.y > si) || (v.y == si && (jj + 1) < i);
        rank += (v.z > si) || (v.z == si && (jj + 2) < i);
        rank += (v.w > si) || (v.w == si && (jj + 3) < i);
      }
    }
    __syncthreads();                             // buffer free before re-issue
  }

  if (act && rank < K) {                         // ranks are a permutation: all K slots filled
    topScores[rank] = si;
    topBoxes[rank * 4 + 0] = boxesAll[i * 4 + 0];
    topBoxes[rank * 4 + 1] = boxesAll[i * 4 + 1];
    topBoxes[rank * 4 + 2] = boxesAll[i * 4 + 2];
    topBoxes[rank * 4 + 3] = boxesAll[i * 4 + 3];
  }
}

// ---------------------------------------------------------------------------
// Kernel 3: greedy NMS over the sorted top-K, fully LDS-resident (needs the
// CDNA5 320 KB LDS: 6000 float4 boxes + areas + alive flags = 144 KB).
// The 96 KB box cache is filled by ONE Tensor-Data-Mover DMA (TENSORcnt)
// issued by wave 0 while all waves compute alive flags from topScores.
// Then serial compaction into the (POST,5) output (col 0 = 0, unused rows 0).
// ---------------------------------------------------------------------------
__global__ __launch_bounds__(1024)
void nms_kernel(const float* __restrict__ topScores,
                const float* __restrict__ topBoxes,
                float* __restrict__ out) {
  __shared__ float4 box[K];        // 96000 B
  __shared__ float  area[K];       // 24000 B
  __shared__ int    alive[K];      // 24000 B

  const int tid = threadIdx.x;

  // Wave 0 kicks off the TDM DMA of all K boxes (K*4 dwords) into LDS.
  if (tid < 32) {                                   // wave-uniform branch
    tdmLoad1D(ldsOffset(&box[0]), topBoxes, (unsigned)(K * 4));
  }

  // Overlap with the DMA: alive flags from global topScores.
  for (int j = tid; j < K; j += 1024) {
    const float s = topScores[j];
    alive[j] = (s > -INFINITY) && (s < INFINITY);   // isfinite (no NaNs upstream)
  }

  if (tid < 32) {
    __builtin_amdgcn_s_wait_tensorcnt(0);           // DMA complete (wave 0)
  }
  __syncthreads();                                  // publish LDS to all waves

  for (int j = tid; j < K; j += 1024) {
    const float4 b = box[j];
    area[j] = (b.z - b.x + 1.0f) * (b.w - b.y + 1.0f);
  }
  __syncthreads();

  // Greedy suppression: iteration i suppresses only if alive[i] still set,
  // exactly matching the reference fori_loop.
  for (int i = 0; i < K; ++i) {
    if (alive[i]) {                                 // uniform LDS broadcast
      const float4 bi = box[i];
      const float  ai = area[i];
      for (int j = i + 1 + tid; j < K; j += 1024) {
        if (alive[j]) {
          const float4 bj = box[j];
          const float iw = fmaxf(fminf(bi.z, bj.z) - fmaxf(bi.x, bj.x) + 1.0f, 0.0f);
          const float ih = fmaxf(fminf(bi.w, bj.w) - fmaxf(bi.y, bj.y) + 1.0f, 0.0f);
          const float inter = iw * ih;
          const float iou = inter / (ai + area[j] - inter);
          if (iou > NMS_THRESH) alive[j] = 0;
        }
      }
    }
    __syncthreads();
  }

  // Output: zero all (POST,5) floats, then pack survivors in rank order.
  for (int t = tid; t < POST * 5; t += 1024) out[t] = 0.0f;
  __syncthreads();
  if (tid == 0) {
    int cnt = 0;
    for (int i = 0; i < K && cnt < POST; ++i) {
      if (alive[i]) {
        const float4 b = box[i];
        out[cnt * 5 + 1] = b.x;
        out[cnt * 5 + 2] = b.y;
        out[cnt * 5 + 3] = b.z;
        out[cnt * 5 + 4] = b.w;
        ++cnt;
      }
    }
  }
}

// ---------------------------------------------------------------------------
extern "C" void kernel_launch(void* const* d_in, const int* in_sizes, int n_in,
                              void* d_out, int out_size, void* d_ws, size_t ws_size,
                              hipStream_t stream) {
  (void)in_sizes; (void)n_in; (void)out_size; (void)ws_size;
  const float* scores  = (const float*)d_in[0];   // (1,18,50,80)
  const float* deltas  = (const float*)d_in[1];   // (1,36,50,80)
  const float* im_info = (const float*)d_in[2];   // (1,3)
  float* out = (float*)d_out;                     // (300,5)

  // Workspace layout (floats): ~850 KB total.
  float* ws        = (float*)d_ws;
  float* boxesAll  = ws;                 // N*4   = 144000
  float* scoresAll = ws + 144000;        // NP    =  36864 (pad = -inf)
  float* topBoxes  = ws + 180864;        // K*4   =  24000 (16B aligned)
  float* topScores = ws + 204864;        // K     =   6000

  decode_kernel<<<NP / 256, 256, 0, stream>>>(scores, deltas, im_info,
                                              boxesAll, scoresAll);
  rank_kernel<<<(N + 255) / 256, 256, 0, stream>>>(scoresAll, boxesAll,
                                                   topScores, topBoxes);
  nms_kernel<<<1, 1024, 0, stream>>>(topScores, topBoxes, out);
}